// GCN_62285615726933
// MI455X (gfx1250) — compile-verified
//
#include <hip/hip_runtime.h>

typedef __attribute__((ext_vector_type(2))) float v2f;
typedef __attribute__((ext_vector_type(8))) float v8f;

// ---------------------------------------------------------------------------
// Degree / normalization
// ---------------------------------------------------------------------------
__global__ __launch_bounds__(256) void deg_init_kernel(float* __restrict__ deg, int n) {
  int i = blockIdx.x * 256 + threadIdx.x;
  if (i < n) deg[i] = 1.0f;  // self-loop contribution
}

__global__ __launch_bounds__(256) void deg_accum_kernel(const int* __restrict__ dst,
                                                        float* __restrict__ deg, int ne) {
  int e = blockIdx.x * 256 + threadIdx.x;
  if (e < ne) unsafeAtomicAdd(&deg[dst[e]], 1.0f);
}

__global__ __launch_bounds__(256) void deg_rsqrt_kernel(float* __restrict__ deg, int n) {
  int i = blockIdx.x * 256 + threadIdx.x;
  if (i < n) deg[i] = rsqrtf(deg[i]);  // deg >= 1 always (self-loops)
}

// ---------------------------------------------------------------------------
// Dense linear: H[N x COLS] = X[N x 128] @ W[128 x COLS] via f32 WMMA 16x16x4.
// One wave computes a 16-row strip across all COLS output columns.
//
// A-frag (16x4, MxK):  lane l -> M = l&15, VGPR0 K = 2*(l>>4), VGPR1 K = +1
// B-frag (4x16, KxN):  lane l -> N = l&15, VGPR0 K = 2*(l>>4), VGPR1 K = +1
// C/D    (16x16)    :  VGPR j, lane l -> M = j + 8*(l>>4), N = l&15
// ---------------------------------------------------------------------------
template <int COLS>
__global__ __launch_bounds__(128) void lin_wmma_kernel(const float* __restrict__ X,
                                                       const float* __restrict__ W,
                                                       float* __restrict__ H, int nrows) {
  constexpr int NT = COLS / 16;
  int wave = blockIdx.x * (blockDim.x >> 5) + (threadIdx.x >> 5);
  int lane = threadIdx.x & 31;
  int mbase = wave * 16;
  if (mbase >= nrows) return;  // wave-uniform exit; EXEC stays all-1s for WMMA

  int mrow = lane & 15;
  int khalf = (lane >> 4) * 2;  // 0 or 2

  v8f acc[NT];
#pragma unroll
  for (int t = 0; t < NT; ++t) acc[t] = (v8f)0.0f;

  const float* __restrict__ xrow = X + (size_t)(mbase + mrow) * 128;

  for (int k = 0; k < 128; k += 4) {
    v2f a;
    a.x = xrow[k + khalf + 0];
    a.y = xrow[k + khalf + 1];
#pragma unroll
    for (int t = 0; t < NT; ++t) {
      int ncol = t * 16 + (lane & 15);
      v2f b;
      b.x = W[(size_t)(k + khalf + 0) * COLS + ncol];
      b.y = W[(size_t)(k + khalf + 1) * COLS + ncol];
      acc[t] = __builtin_amdgcn_wmma_f32_16x16x4_f32(
          /*neg_a=*/false, a, /*neg_b=*/false, b,
          /*c_mod=*/(short)0, acc[t], /*reuse_a=*/false, /*reuse_b=*/false);
    }
  }

  int nlo = lane & 15;
  int rowoff = (lane >> 4) * 8;
#pragma unroll
  for (int t = 0; t < NT; ++t) {
#pragma unroll
    for (int j = 0; j < 8; ++j) {
      H[(size_t)(mbase + rowoff + j) * COLS + t * 16 + nlo] = acc[t][j];
    }
  }
}

// ---------------------------------------------------------------------------
// Edge scatter: one wave per edge, each lane owns VEC channels.
// agg[dst] += h[src] * (dinv[src]*dinv[dst])   via global_atomic_add_f32.
// ---------------------------------------------------------------------------
template <int VEC>  // channels = 32*VEC
__global__ __launch_bounds__(256) void edge_scatter_kernel(const int* __restrict__ src,
                                                           const int* __restrict__ dst,
                                                           const float* __restrict__ dinv,
                                                           const float* __restrict__ h,
                                                           float* __restrict__ agg, int ne) {
  int gid = blockIdx.x * 256 + threadIdx.x;
  int e = gid >> 5;
  int lane = gid & 31;
  if (e >= ne) return;
  int s = src[e];
  int d = dst[e];
  float nrm = dinv[s] * dinv[d];
  const float* __restrict__ hp = h + (size_t)s * (32 * VEC) + lane * VEC;
  float* ap = agg + (size_t)d * (32 * VEC) + lane * VEC;
  if constexpr (VEC == 4) {
    float4 v = *(const float4*)hp;
    unsafeAtomicAdd(ap + 0, v.x * nrm);
    unsafeAtomicAdd(ap + 1, v.y * nrm);
    unsafeAtomicAdd(ap + 2, v.z * nrm);
    unsafeAtomicAdd(ap + 3, v.w * nrm);
  } else {
    float2 v = *(const float2*)hp;
    unsafeAtomicAdd(ap + 0, v.x * nrm);
    unsafeAtomicAdd(ap + 1, v.y * nrm);
  }
}

// ---------------------------------------------------------------------------
// Finalize: out = agg + h*dinv^2 (self loop) + bias, optional ReLU. float4 lanes.
// ---------------------------------------------------------------------------
template <int C, bool RELU>
__global__ __launch_bounds__(256) void finalize_kernel(const float* __restrict__ agg,
                                                       const float* __restrict__ h,
                                                       const float* __restrict__ dinv,
                                                       const float* __restrict__ bias,
                                                       float* __restrict__ out, int n) {
  constexpr int R = C / 4;  // float4 per row
  int idx = blockIdx.x * 256 + threadIdx.x;
  if (idx >= n * R) return;
  int i = idx / R;
  int c4 = (idx - i * R) * 4;
  float di = dinv[i];
  float d2 = di * di;
  float4 a = ((const float4*)agg)[idx];
  float4 hv = ((const float4*)h)[idx];
  float4 r;
  r.x = fmaf(hv.x, d2, a.x) + bias[c4 + 0];
  r.y = fmaf(hv.y, d2, a.y) + bias[c4 + 1];
  r.z = fmaf(hv.z, d2, a.z) + bias[c4 + 2];
  r.w = fmaf(hv.w, d2, a.w) + bias[c4 + 3];
  if (RELU) {
    r.x = fmaxf(r.x, 0.0f);
    r.y = fmaxf(r.y, 0.0f);
    r.z = fmaxf(r.z, 0.0f);
    r.w = fmaxf(r.w, 0.0f);
  }
  ((float4*)out)[idx] = r;
}

// ---------------------------------------------------------------------------
// Final linear: out[N x 2] = h3[N x 64] @ Wout[64 x 2] + bout
// ---------------------------------------------------------------------------
__global__ __launch_bounds__(256) void out_linear_kernel(const float* __restrict__ h,
                                                         const float* __restrict__ Wout,
                                                         const float* __restrict__ bout,
                                                         float* __restrict__ out, int n) {
  int i = blockIdx.x * 256 + threadIdx.x;
  if (i >= n) return;
  const float* __restrict__ hr = h + (size_t)i * 64;
  float a0 = bout[0], a1 = bout[1];
#pragma unroll 8
  for (int c = 0; c < 64; ++c) {
    float v = hr[c];
    a0 = fmaf(v, Wout[c * 2 + 0], a0);
    a1 = fmaf(v, Wout[c * 2 + 1], a1);
  }
  out[(size_t)i * 2 + 0] = a0;
  out[(size_t)i * 2 + 1] = a1;
}

// ---------------------------------------------------------------------------
// Orchestration
// ---------------------------------------------------------------------------
extern "C" void kernel_launch(void* const* d_in, const int* in_sizes, int n_in,
                              void* d_out, int out_size, void* d_ws, size_t ws_size,
                              hipStream_t stream) {
  const float* x    = (const float*)d_in[0];
  const int*   ei   = (const int*)d_in[1];
  const float* W1   = (const float*)d_in[2];
  const float* b1   = (const float*)d_in[3];
  const float* W2   = (const float*)d_in[4];
  const float* b2   = (const float*)d_in[5];
  const float* W3   = (const float*)d_in[6];
  const float* b3   = (const float*)d_in[7];
  const float* Wout = (const float*)d_in[8];
  const float* bout = (const float*)d_in[9];

  const int N = in_sizes[0] / 128;
  const int E = in_sizes[1] / 2;
  const int* src = ei;       // edge_index[0]
  const int* dst = ei + E;   // edge_index[1]

  float* ws   = (float*)d_ws;
  float* dinv = ws;                               // N
  float* bufH = ws + N;                           // N*128 (linear output)
  float* bufA = bufH + (size_t)N * 128;           // N*128 (agg / activation)

  float* outp = (float*)d_out;                    // [N,2]
  float* hout = outp + (size_t)N * 2;             // [N,64] embedding output

  const int B = 256;
  dim3 blk(B);
  int nb_nodes   = (N + B - 1) / B;
  int nb_edges   = (E + B - 1) / B;
  long long sth  = (long long)E * 32;
  int nb_scatter = (int)((sth + B - 1) / B);
  int waves_m    = (N + 15) / 16;
  int nb_gemm    = (waves_m + 3) / 4;             // 4 waves / 128-thread block
  int nb_fin128  = ((N * 32) + B - 1) / B;        // N*128/4 float4 elems
  int nb_fin64   = ((N * 16) + B - 1) / B;

  // --- normalization: deg -> dinv ---
  deg_init_kernel<<<nb_nodes, blk, 0, stream>>>(dinv, N);
  deg_accum_kernel<<<nb_edges, blk, 0, stream>>>(dst, dinv, E);
  deg_rsqrt_kernel<<<nb_nodes, blk, 0, stream>>>(dinv, N);

  // --- conv1: x @ W1 -> scatter -> +self +b1, ReLU -> bufA ---
  lin_wmma_kernel<128><<<nb_gemm, 128, 0, stream>>>(x, W1, bufH, N);
  hipMemsetAsync(bufA, 0, (size_t)N * 128 * sizeof(float), stream);
  edge_scatter_kernel<4><<<nb_scatter, blk, 0, stream>>>(src, dst, dinv, bufH, bufA, E);
  finalize_kernel<128, true><<<nb_fin128, blk, 0, stream>>>(bufA, bufH, dinv, b1, bufA, N);

  // --- conv2: bufA @ W2 -> scatter -> +self +b2, ReLU -> bufA ---
  lin_wmma_kernel<128><<<nb_gemm, 128, 0, stream>>>(bufA, W2, bufH, N);
  hipMemsetAsync(bufA, 0, (size_t)N * 128 * sizeof(float), stream);
  edge_scatter_kernel<4><<<nb_scatter, blk, 0, stream>>>(src, dst, dinv, bufH, bufA, E);
  finalize_kernel<128, true><<<nb_fin128, blk, 0, stream>>>(bufA, bufH, dinv, b2, bufA, N);

  // --- conv3 (embedding, 64ch, no ReLU): bufA @ W3 -> scatter -> hout ---
  lin_wmma_kernel<64><<<nb_gemm, 128, 0, stream>>>(bufA, W3, bufH, N);
  hipMemsetAsync(hout, 0, (size_t)N * 64 * sizeof(float), stream);
  edge_scatter_kernel<2><<<nb_scatter, blk, 0, stream>>>(src, dst, dinv, bufH, hout, E);
  finalize_kernel<64, false><<<nb_fin64, blk, 0, stream>>>(hout, bufH, dinv, b3, hout, N);

  // --- final linear: hout @ Wout + bout -> outp ---
  out_linear_kernel<<<nb_nodes, blk, 0, stream>>>(hout, Wout, bout, outp, N);
}